// PCAVolume_67997922230551
// MI455X (gfx1250) — compile-verified
//
#include <hip/hip_runtime.h>
#include <hip/hip_bf16.h>
#include <stdint.h>

// ---------------------------------------------------------------------------
// CDNA5-native scatter atomics (single-instruction f32 RMW at the L2 atomic
// units; no CAS loops). GVS addressing: uniform SGPR-pair base + VGPR byte
// offset + compile-time immediate offset folded into the 24-bit IOFFSET.
// ---------------------------------------------------------------------------
template <int IMM>
__device__ __forceinline__ void gatomic_add_f32(float* base, unsigned byte_off, float v) {
  asm volatile("global_atomic_add_f32 %0, %1, %2 offset:%3 scope:SCOPE_DEV"
               :: "v"(byte_off), "v"(v), "s"(base), "i"(IMM) : "memory");
}
template <int IMM>
__device__ __forceinline__ void gatomic_max_f32(float* base, unsigned byte_off, float v) {
  asm volatile("global_atomic_max_num_f32 %0, %1, %2 offset:%3 scope:SCOPE_DEV"
               :: "v"(byte_off), "v"(v), "s"(base), "i"(IMM) : "memory");
}
template <int IMM>
__device__ __forceinline__ void gatomic_min_f32(float* base, unsigned byte_off, float v) {
  asm volatile("global_atomic_min_num_f32 %0, %1, %2 offset:%3 scope:SCOPE_DEV"
               :: "v"(byte_off), "v"(v), "s"(base), "i"(IMM) : "memory");
}

// ---------------------------------------------------------------------------
// CDNA5 async global->LDS gather (ASYNCcnt-tracked), 16B per lane.
// ---------------------------------------------------------------------------
__device__ __forceinline__ void async_gather16(const float4* gbase, unsigned gbyte_off,
                                               void* lds_dst) {
  unsigned lds_off = (unsigned)(unsigned long long)lds_dst;  // low 32 bits = LDS offset
  asm volatile("global_load_async_to_lds_b128 %0, %1, %2"
               :: "v"(lds_off), "v"(gbyte_off), "s"(gbase) : "memory");
}

template <int N>
__device__ __forceinline__ void wait_asynccnt_le() {
#if __has_builtin(__builtin_amdgcn_s_wait_asynccnt)
  __builtin_amdgcn_s_wait_asynccnt(N);
#else
  if (N == 0) asm volatile("s_wait_asynccnt 0" ::: "memory");
  else        asm volatile("s_wait_asynccnt 1" ::: "memory");
#endif
  asm volatile("" ::: "memory");  // compiler barrier: keep LDS reads after wait
}

// ---------------------------------------------------------------------------
// Closed-form 3x3 symmetric eigensolver (trigonometric eigenvalues,
// cross-product eigenvectors). Ascending eigenvalues; columns = eigenvectors.
// NOTE: 3x3 per-voxel matrices cannot map onto 16x16xK WMMA shapes without a
// global edge sort; VALU closed-form is the right tool on CDNA5 here.
// ---------------------------------------------------------------------------
__device__ __forceinline__ float3 cross3(float ax, float ay, float az,
                                         float bx, float by, float bz) {
  return make_float3(ay * bz - az * by, az * bx - ax * bz, ax * by - ay * bx);
}

__device__ __forceinline__ float3 eig_vec(float a00, float a01, float a02,
                                          float a11, float a12, float a22,
                                          float lam, float3 fb) {
  float r0x = a00 - lam, r0y = a01, r0z = a02;
  float r1x = a01, r1y = a11 - lam, r1z = a12;
  float r2x = a02, r2y = a12, r2z = a22 - lam;
  float3 c0 = cross3(r0x, r0y, r0z, r1x, r1y, r1z);
  float3 c1 = cross3(r0x, r0y, r0z, r2x, r2y, r2z);
  float3 c2 = cross3(r1x, r1y, r1z, r2x, r2y, r2z);
  float n0 = c0.x * c0.x + c0.y * c0.y + c0.z * c0.z;
  float n1 = c1.x * c1.x + c1.y * c1.y + c1.z * c1.z;
  float n2 = c2.x * c2.x + c2.y * c2.y + c2.z * c2.z;
  float3 best = c0; float nb = n0;
  if (n1 > nb) { best = c1; nb = n1; }
  if (n2 > nb) { best = c2; nb = n2; }
  if (nb < 1e-30f) return fb;
  float r = rsqrtf(nb);
  return make_float3(best.x * r, best.y * r, best.z * r);
}

__device__ void eig3(float a00, float a01, float a02, float a11, float a12, float a22,
                     float ev[3], float Q[3][3]) {
  float p1 = a01 * a01 + a02 * a02 + a12 * a12;
  float q  = (a00 + a11 + a22) * (1.0f / 3.0f);
  float b00 = a00 - q, b11 = a11 - q, b22 = a22 - q;
  float p2 = b00 * b00 + b11 * b11 + b22 * b22 + 2.0f * p1;
  float p  = sqrtf(fmaxf(p2, 0.0f) * (1.0f / 6.0f));
  float l0, l1, l2;
  if (p < 1e-25f) {
    l0 = l1 = l2 = q;
  } else {
    float pi  = 1.0f / p;
    float c00 = b00 * pi, c11 = b11 * pi, c22 = b22 * pi;
    float c01 = a01 * pi, c02 = a02 * pi, c12 = a12 * pi;
    float detB = c00 * (c11 * c22 - c12 * c12)
               - c01 * (c01 * c22 - c12 * c02)
               + c02 * (c01 * c12 - c11 * c02);
    float r   = fminf(1.0f, fmaxf(-1.0f, 0.5f * detB));
    float phi = acosf(r) * (1.0f / 3.0f);
    float big = q + 2.0f * p * cosf(phi);
    float sml = q + 2.0f * p * cosf(phi + 2.0943951023931953f);  // + 2*pi/3
    float mid = 3.0f * q - big - sml;
    l0 = sml; l1 = mid; l2 = big;
  }
  ev[0] = l0; ev[1] = l1; ev[2] = l2;
  float3 v0 = eig_vec(a00, a01, a02, a11, a12, a22, l0, make_float3(1, 0, 0));
  float3 v2 = eig_vec(a00, a01, a02, a11, a12, a22, l2, make_float3(0, 0, 1));
  float3 v1 = cross3(v2.x, v2.y, v2.z, v0.x, v0.y, v0.z);
  float n1 = v1.x * v1.x + v1.y * v1.y + v1.z * v1.z;
  if (n1 < 1e-30f) v1 = make_float3(0, 1, 0);
  else { float r = rsqrtf(n1); v1.x *= r; v1.y *= r; v1.z *= r; }
  float3 v2b = cross3(v0.x, v0.y, v0.z, v1.x, v1.y, v1.z);
  Q[0][0] = v0.x;  Q[1][0] = v0.y;  Q[2][0] = v0.z;
  Q[0][1] = v1.x;  Q[1][1] = v1.y;  Q[2][1] = v1.z;
  Q[0][2] = v2b.x; Q[1][2] = v2b.y; Q[2][2] = v2b.z;
}

// ---------------------------------------------------------------------------
// Pass A: single-sweep moment accumulation. Point gathers are software-
// pipelined through LDS with the CDNA5 async-load path (double buffer,
// ASYNCcnt <= 1 overlap). 11 native f32 atomic adds per edge.
// ws layout (floats): [0,V) cnt | [V,5V) sum_c (SoA, 4 comps) | [5V,11V) S2 (SoA, 6)
// All loop indexing is 32-bit (E < 2^31) to keep address math off the 64-bit path.
// ---------------------------------------------------------------------------
__global__ void __launch_bounds__(256) pca_accum(const float4* __restrict__ pts,
                                                 const int* __restrict__ e_base,
                                                 const int* __restrict__ e_voxel,
                                                 float* __restrict__ ws,
                                                 int E, int V) {
  __shared__ float4 stage[2][256];
  float* cntp = ws;
  float* sump = ws + V;
  float* s2p  = ws + 5 * (size_t)V;
  const int tid = threadIdx.x;
  const int stride = (int)(gridDim.x * blockDim.x);
  const int i0 = (int)(blockIdx.x * blockDim.x) + tid;

  if (i0 < E) {
    unsigned goff = (unsigned)e_base[i0] * 16u;
    async_gather16(pts, goff, &stage[0][tid]);
  }
  int buf = 0;
  for (int i = i0; i < E; i += stride) {
    int inx = i + stride;
    if (inx < E) {
      unsigned goff = (unsigned)e_base[inx] * 16u;
      async_gather16(pts, goff, &stage[buf ^ 1][tid]);
      wait_asynccnt_le<1>();   // current buffer's (older) async load is done
    } else {
      wait_asynccnt_le<0>();
    }
    float4 p = stage[buf][tid];
    buf ^= 1;

    unsigned off = (unsigned)e_voxel[i] * 4u;
    float dx = p.y, dy = p.z, dz = p.w;  // xyz = components 1..3
    gatomic_add_f32<0>(cntp, off, 1.0f);
    gatomic_add_f32<0>(sump + 0 * (size_t)V, off, p.x);
    gatomic_add_f32<0>(sump + 1 * (size_t)V, off, p.y);
    gatomic_add_f32<0>(sump + 2 * (size_t)V, off, p.z);
    gatomic_add_f32<0>(sump + 3 * (size_t)V, off, p.w);
    gatomic_add_f32<0>(s2p + 0 * (size_t)V, off, dx * dx);
    gatomic_add_f32<0>(s2p + 1 * (size_t)V, off, dx * dy);
    gatomic_add_f32<0>(s2p + 2 * (size_t)V, off, dx * dz);
    gatomic_add_f32<0>(s2p + 3 * (size_t)V, off, dy * dy);
    gatomic_add_f32<0>(s2p + 4 * (size_t)V, off, dy * dz);
    gatomic_add_f32<0>(s2p + 5 * (size_t)V, off, dz * dz);
  }
}

// ---------------------------------------------------------------------------
// Pass B: per-voxel finalize. cov = S2/n - mu*mu^T (== mean centered ddT).
// Writes bxyz, volume, mask, eigvals, eigvecs; inits seg-max/min to -inf/+inf.
// out layout (floats): bxyz[4V] | vol[V] | mask[V] | eval[3V] | evec[9V] |
//                      lmax[3V] | lmin[3V]
// ---------------------------------------------------------------------------
__global__ void __launch_bounds__(256) pca_voxel(const float4* __restrict__ bcenter,
                                                 const float* __restrict__ ws,
                                                 float* __restrict__ out, int V) {
  int v = blockIdx.x * blockDim.x + threadIdx.x;
  if (v >= V) return;
  const size_t Vz = (size_t)V;
  float cnt = ws[v];
  float s0 = ws[Vz + v], s1 = ws[2 * Vz + v], s2c = ws[3 * Vz + v], s3 = ws[4 * Vz + v];
  float xx = ws[5 * Vz + v], xy = ws[6 * Vz + v], xz = ws[7 * Vz + v];
  float yy = ws[8 * Vz + v], yz = ws[9 * Vz + v], zz = ws[10 * Vz + v];

  bool mask = cnt > 0.5f;
  float inv = 1.0f / fmaxf(cnt, 1.0f);
  float4 bc = bcenter[v];
  float m0 = mask ? s0 * inv : bc.x;
  float m1 = mask ? s1 * inv : bc.y;
  float m2 = mask ? s2c * inv : bc.z;
  float m3 = mask ? s3 * inv : bc.w;

  float a00 = 0, a01 = 0, a02 = 0, a11 = 0, a12 = 0, a22 = 0;
  if (mask) {
    float mu1 = s1 * inv, mu2 = s2c * inv, mu3 = s3 * inv;
    a00 = xx * inv - mu1 * mu1; a01 = xy * inv - mu1 * mu2; a02 = xz * inv - mu1 * mu3;
    a11 = yy * inv - mu2 * mu2; a12 = yz * inv - mu2 * mu3; a22 = zz * inv - mu3 * mu3;
  }
  float ev[3]; float Q[3][3];
  eig3(a00, a01, a02, a11, a12, a22, ev, Q);

  ((float4*)out)[v] = make_float4(m0, m1, m2, m3);
  out[4 * Vz + v] = cnt;
  out[5 * Vz + v] = mask ? 1.0f : 0.0f;
  float* eval = out + 6 * Vz;
  float* evec = out + 9 * Vz;
  float* lmax = out + 18 * Vz;
  float* lmin = out + 21 * Vz;
  eval[3 * v + 0] = ev[0]; eval[3 * v + 1] = ev[1]; eval[3 * v + 2] = ev[2];
#pragma unroll
  for (int i = 0; i < 3; ++i)
#pragma unroll
    for (int j = 0; j < 3; ++j) evec[9 * (size_t)v + 3 * i + j] = Q[i][j];
  const float INF = __builtin_inff();
  lmax[3 * v + 0] = -INF; lmax[3 * v + 1] = -INF; lmax[3 * v + 2] = -INF;
  lmin[3 * v + 0] =  INF; lmin[3 * v + 1] =  INF; lmin[3 * v + 2] =  INF;
}

// ---------------------------------------------------------------------------
// Pass C: per-edge eigenbasis projection + native f32 atomic max/min with
// immediate offsets (one VGPR offset serves all 3 components per buffer).
// bxyz and evec gathers hit L2 (both arrays fully L2-resident on MI455X).
// ---------------------------------------------------------------------------
__global__ void __launch_bounds__(256) pca_proj(const float4* __restrict__ pts,
                                                const int* __restrict__ e_base,
                                                const int* __restrict__ e_voxel,
                                                float* __restrict__ out,
                                                int E, int V) {
  const size_t Vz = (size_t)V;
  const float4* bxyz = (const float4*)out;
  const float* evec = out + 9 * Vz;
  float* lmax = out + 18 * Vz;
  float* lmin = out + 21 * Vz;
  const int stride = (int)(gridDim.x * blockDim.x);
  for (int i = (int)(blockIdx.x * blockDim.x) + (int)threadIdx.x; i < E; i += stride) {
    int b = e_base[i];
    int vx = e_voxel[i];
    float4 p = pts[b];
    float4 c = bxyz[vx];
    float d1 = p.y - c.y, d2 = p.z - c.z, d3 = p.w - c.w;
    const float* E9 = evec + 9 * (size_t)vx;
    float e0 = E9[0], e1 = E9[1], e2 = E9[2];
    float e3 = E9[3], e4 = E9[4], e5 = E9[5];
    float e6 = E9[6], e7 = E9[7], e8 = E9[8];
    float pr0 = d1 * e0 + d2 * e3 + d3 * e6;
    float pr1 = d1 * e1 + d2 * e4 + d3 * e7;
    float pr2 = d1 * e2 + d2 * e5 + d3 * e8;
    unsigned off = (unsigned)vx * 12u;
    gatomic_max_f32<0>(lmax, off, pr0);
    gatomic_max_f32<4>(lmax, off, pr1);
    gatomic_max_f32<8>(lmax, off, pr2);
    gatomic_min_f32<0>(lmin, off, pr0);
    gatomic_min_f32<4>(lmin, off, pr1);
    gatomic_min_f32<8>(lmin, off, pr2);
  }
}

// Pass D: empty voxels get 0 in seg-max/min (torch_scatter fill semantics).
__global__ void __launch_bounds__(256) pca_fix(float* __restrict__ out, int V) {
  int v = blockIdx.x * blockDim.x + threadIdx.x;
  if (v >= V) return;
  const size_t Vz = (size_t)V;
  if (out[4 * Vz + v] <= 0.5f) {
#pragma unroll
    for (int j = 0; j < 3; ++j) {
      out[18 * Vz + 3 * (size_t)v + j] = 0.0f;
      out[21 * Vz + 3 * (size_t)v + j] = 0.0f;
    }
  }
}

extern "C" void kernel_launch(void* const* d_in, const int* in_sizes, int n_in,
                              void* d_out, int out_size, void* d_ws, size_t ws_size,
                              hipStream_t stream) {
  const float4* pts     = (const float4*)d_in[0];
  const float4* bcenter = (const float4*)d_in[1];
  const int* e_base     = (const int*)d_in[2];
  const int* e_voxel    = (const int*)d_in[3];
  const int E = in_sizes[2];
  const int V = in_sizes[1] / 4;
  float* ws  = (float*)d_ws;
  float* out = (float*)d_out;

  hipMemsetAsync(ws, 0, (size_t)11 * (size_t)V * sizeof(float), stream);

  const int blk = 256;           // 8 waves of 32 on CDNA5
  const int edge_grid = 4096;    // divides E=8.4M exactly -> uniform 8-deep loops
  pca_accum<<<edge_grid, blk, 0, stream>>>(pts, e_base, e_voxel, ws, E, V);
  pca_voxel<<<(V + blk - 1) / blk, blk, 0, stream>>>(bcenter, ws, out, V);
  pca_proj<<<edge_grid, blk, 0, stream>>>(pts, e_base, e_voxel, out, E, V);
  pca_fix<<<(V + blk - 1) / blk, blk, 0, stream>>>(out, V);
}